// SIDARTHEOde_66838281060861
// MI455X (gfx1250) — compile-verified
//
#include <hip/hip_runtime.h>

// SIDARTHE RK4, 100 steps, 8-state. Single wave32; linear part of f(y)
// evaluated with chained V_WMMA_F32_16X16X4_F32 (cols 0-3, then 4-7 of the
// 8x8 operator L). L's rows are duplicated into rows 8-15 so the 16x16 f32
// C/D layout leaves L*z replicated in every lane -> no cross-lane ops.

typedef __attribute__((ext_vector_type(2))) float v2f;
typedef __attribute__((ext_vector_type(8))) float v8f;

#define LEN_DATA 100

__global__ __launch_bounds__(32) void sidarthe_rk4_wmma(
    const float* __restrict__ I0, const float* __restrict__ E0,
    const float* __restrict__ H0, const float* __restrict__ rates,
    const float* __restrict__ states, float* __restrict__ out)
{
    const int lane = threadIdx.x;           // single wave32
    __shared__ float Lm[64];                // 8x8 row-major operator

    // rates (uniform scalar loads)
    const float alpha = rates[0],  beta  = rates[1],  gamma = rates[2],  delta = rates[3];
    const float eps_  = rates[4],  theta = rates[5],  zeta  = rates[6],  eta   = rates[7];
    const float mu    = rates[8],  nu    = rates[9],  tau   = rates[10], lam   = rates[11];
    const float kappa = rates[12], xi    = rates[13], rho   = rates[14], sigma = rates[15];

    // Build L in LDS (once; hot loop never touches memory)
    Lm[lane] = 0.0f; Lm[lane + 32] = 0.0f;
    __syncthreads();
    if (lane == 0) {
        // state order: [S, I, D, A, R, T, H, E]
        Lm[1*8+1] = -(eps_ + zeta + lam);
        Lm[2*8+1] = eps_;   Lm[2*8+2] = -(eta + rho);
        Lm[3*8+1] = zeta;   Lm[3*8+3] = -(theta + mu + kappa);
        Lm[4*8+2] = eta;    Lm[4*8+3] = theta;   Lm[4*8+4] = -(nu + xi);
        Lm[5*8+3] = mu;     Lm[5*8+4] = nu;      Lm[5*8+5] = -(sigma + tau);
        Lm[6*8+1] = lam;    Lm[6*8+2] = rho;     Lm[6*8+3] = kappa;
        Lm[6*8+4] = xi;     Lm[6*8+5] = sigma;
        Lm[7*8+5] = tau;
    }
    __syncthreads();

    // A-operand (16x4 f32): lane = row M (rows 8-15 duplicate rows 0-7),
    // lane-half selects K pair {0,1} vs {2,3}.
    const int  row   = lane & 7;
    const int  khalf = (lane >> 4) & 1;     // 0: K={0,1}/{4,5}; 1: K={2,3}/{6,7}
    v2f A_lo, A_hi;
    A_lo[0] = Lm[row*8 + 2*khalf + 0];
    A_lo[1] = Lm[row*8 + 2*khalf + 1];
    A_hi[0] = Lm[row*8 + 4 + 2*khalf + 0];
    A_hi[1] = Lm[row*8 + 4 + 2*khalf + 1];

    const bool hiHalf = (lane >= 16);

    // Replicated state on every lane: y = [S, I, D, A, R, T, H, E]
    float y[8];
    y[0] = states[0]; y[1] = I0[0];     y[2] = states[1]; y[3] = states[2];
    y[4] = states[3]; y[5] = states[4]; y[6] = H0[0];     y[7] = E0[0];

    // f(z) = L*z + infection*e01 : two chained WMMAs + 5 scalar FMAs
    auto feval = [&](const float* z, float* k) {
        v2f b_lo, b_hi;                     // B (4x16): z broadcast across N
        b_lo[0] = hiHalf ? z[2] : z[0];
        b_lo[1] = hiHalf ? z[3] : z[1];
        b_hi[0] = hiHalf ? z[6] : z[4];
        b_hi[1] = hiHalf ? z[7] : z[5];
        v8f acc = {};
        acc = __builtin_amdgcn_wmma_f32_16x16x4_f32(false, A_lo, false, b_lo,
                                                    (short)0, acc, false, false);
        acc = __builtin_amdgcn_wmma_f32_16x16x4_f32(false, A_hi, false, b_hi,
                                                    (short)0, acc, false, false);
        const float inf = z[0] * (alpha*z[1] + beta*z[2] + gamma*z[3] + delta*z[4]);
#pragma unroll
        for (int j = 0; j < 8; ++j) k[j] = acc[j];
        k[0] += inf;
        k[1] += inf;
    };

    if (lane == 0) {
#pragma unroll
        for (int j = 0; j < 8; ++j) out[j] = y[j];   // row 0 = y0
    }

    const float dt = 1.0f;
    for (int step = 1; step <= LEN_DATA; ++step) {
        float k1[8], k2[8], k3[8], k4[8], z[8];
        feval(y, k1);
#pragma unroll
        for (int j = 0; j < 8; ++j) z[j] = y[j] + 0.5f*dt*k1[j];
        feval(z, k2);
#pragma unroll
        for (int j = 0; j < 8; ++j) z[j] = y[j] + 0.5f*dt*k2[j];
        feval(z, k3);
#pragma unroll
        for (int j = 0; j < 8; ++j) z[j] = y[j] + dt*k3[j];
        feval(z, k4);
#pragma unroll
        for (int j = 0; j < 8; ++j)
            y[j] += (dt/6.0f) * (k1[j] + 2.0f*k2[j] + 2.0f*k3[j] + k4[j]);

        if (lane == 0) {                    // fire-and-forget row store
            float* o = out + 8*step;
#pragma unroll
            for (int j = 0; j < 8; ++j) o[j] = y[j];
        }
    }
}

extern "C" void kernel_launch(void* const* d_in, const int* in_sizes, int n_in,
                              void* d_out, int out_size, void* d_ws, size_t ws_size,
                              hipStream_t stream) {
    (void)in_sizes; (void)n_in; (void)out_size; (void)d_ws; (void)ws_size;
    const float* I0     = (const float*)d_in[0];
    const float* E0     = (const float*)d_in[1];
    const float* H0     = (const float*)d_in[2];
    const float* rates  = (const float*)d_in[3];
    const float* states = (const float*)d_in[4];
    sidarthe_rk4_wmma<<<1, 32, 0, stream>>>(I0, E0, H0, rates, states, (float*)d_out);
}